// MovieExpertCRS_49598282334489
// MI455X (gfx1250) — compile-verified
//
#include <hip/hip_runtime.h>
#include <math.h>

// ---- problem constants (match reference) ----
#define N_ENTITY 30000
#define N_REL    18
#define N_BASES  8
#define DMODEL   128
#define NEDGE    1000000
#define BATCH    64
#define SEQL     256
#define NCTX     32
#define FFND     512
#define NEGF     (-1e9f)

typedef __attribute__((ext_vector_type(2))) float v2f;
typedef __attribute__((ext_vector_type(8))) float v8f;

// ---------------------------------------------------------------- utilities
__global__ void mek_zero(float* __restrict__ p, long n) {
  long i = (long)blockIdx.x * blockDim.x + threadIdx.x;
  long stride = (long)gridDim.x * blockDim.x;
  for (; i < n; i += stride) p[i] = 0.f;
}

// ---------------------------------------------------------------- RGCN
__global__ void mek_edge_cnt(const int* __restrict__ ei, const int* __restrict__ et,
                             float* __restrict__ cnt) {
  int e = blockIdx.x * 256 + threadIdx.x;
  if (e >= NEDGE) return;
  int dst = ei[NEDGE + e];
  atomicAdd(&cnt[dst * N_REL + et[e]], 1.f);
}

// one wave32 per edge; lane covers 4 consecutive d.  basis (122.9MB) is
// L2-resident on MI455X (192MB L2), so these gathers stay on-chip.
__global__ void mek_edge_msg(const int* __restrict__ ei, const int* __restrict__ et,
                             const float* __restrict__ basis, const float* __restrict__ comp,
                             const float* __restrict__ cnt, float* __restrict__ agg) {
  int wave = threadIdx.x >> 5, lane = threadIdx.x & 31;
  int e = blockIdx.x * 8 + wave;
  if (e >= NEDGE) return;
  int src = ei[e], dst = ei[NEDGE + e], r = et[e];
  float4 acc = make_float4(0.f, 0.f, 0.f, 0.f);
  for (int bb = 0; bb < N_BASES; ++bb) {
    float cb = comp[r * N_BASES + bb];
    const float4* bp = (const float4*)(basis + ((size_t)bb * N_ENTITY + src) * DMODEL);
    float4 t = bp[lane];
    acc.x += cb * t.x; acc.y += cb * t.y; acc.z += cb * t.z; acc.w += cb * t.w;
  }
  float nrm = 1.f / fmaxf(cnt[dst * N_REL + r], 1.f);
  float* op = agg + (size_t)dst * DMODEL + lane * 4;
  atomicAdd(op + 0, acc.x * nrm);
  atomicAdd(op + 1, acc.y * nrm);
  atomicAdd(op + 2, acc.z * nrm);
  atomicAdd(op + 3, acc.w * nrm);
}

__global__ void mek_kg_fin(float* __restrict__ kg, const float* __restrict__ root,
                           const float* __restrict__ bias) {
  long i = (long)blockIdx.x * 256 + threadIdx.x;
  if (i >= (long)N_ENTITY * DMODEL) return;
  kg[i] += root[i] + bias[i & (DMODEL - 1)];
}

// ---------------------------------------------------------------- generic f32 WMMA GEMM
// C[M,N] = A[M,K] * B[K,N] (+bias[N]) (+addsrc) (relu).  One 16x16 C tile per
// wave using V_WMMA_F32_16X16X4_F32.  Wave-uniform guard -> EXEC stays all-1s.
__global__ void mek_gemm(const float* __restrict__ A, const float* __restrict__ Bm,
                         float* __restrict__ C, const float* __restrict__ bias,
                         const float* __restrict__ addsrc,
                         int M, int N, int K, int lda, int ldb, int ldc, int relu) {
  int wave = threadIdx.x >> 5, lane = threadIdx.x & 31;
  int tilesN = N >> 4;
  int total = (M >> 4) * tilesN;
  int tile = blockIdx.x * 8 + wave;
  if (tile >= total) return;
  int mt = tile / tilesN, nt = tile % tilesN;
  int m = lane & 15, kh = lane >> 4;
  v8f c = {};
  const float* ap  = A  + (size_t)(mt * 16 + m) * lda + 2 * kh;
  const float* bp0 = Bm + (size_t)(2 * kh) * ldb + nt * 16 + m;
  for (int kk = 0; kk < K; kk += 4) {
    v2f a, bf;
    a[0] = ap[kk];
    a[1] = ap[kk + 1];
    const float* bp = bp0 + (size_t)kk * ldb;
    bf[0] = bp[0];
    bf[1] = bp[ldb];
    c = __builtin_amdgcn_wmma_f32_16x16x4_f32(false, a, false, bf, (short)0, c, false, false);
  }
  int col = nt * 16 + m;
  for (int r2 = 0; r2 < 8; ++r2) {
    int row = mt * 16 + r2 + 8 * kh;
    float val = c[r2];
    if (bias)   val += bias[col];
    if (addsrc) val += addsrc[(size_t)row * ldc + col];
    if (relu)   val = fmaxf(val, 0.f);
    C[(size_t)row * ldc + col] = val;
  }
}

// ---------------------------------------------------------------- encoder pieces
__global__ void mek_embed(const int* __restrict__ tok, const float* __restrict__ te,
                          const float* __restrict__ pe, float* __restrict__ x) {
  long i = (long)blockIdx.x * 256 + threadIdx.x;
  if (i >= (long)BATCH * SEQL * DMODEL) return;
  int d = (int)(i & (DMODEL - 1));
  long row = i >> 7;             // b*SEQL + l
  int l = (int)(row & (SEQL - 1));
  int t = tok[row];
  x[i] = te[(size_t)t * DMODEL + d] * 11.3137085f + pe[l * DMODEL + d];
}

// per (b, h, q-row) block: scores, softmax, a@v
__global__ void mek_attn(const float* __restrict__ q, const float* __restrict__ k,
                         const float* __restrict__ v, const int* __restrict__ tok,
                         float* __restrict__ ao) {
  int blk = blockIdx.x;
  int qr = blk & (SEQL - 1);
  int h  = (blk >> 8) & 1;
  int b  = blk >> 9;
  int tid = threadIdx.x;          // 256
  __shared__ float ss[SEQL];
  __shared__ float red[SEQL];
  const float* qp = q + ((size_t)(b * SEQL + qr)) * DMODEL + h * 64;
  const float* kp = k + ((size_t)(b * SEQL + tid)) * DMODEL + h * 64;
  float s = 0.f;
  for (int d2 = 0; d2 < 64; ++d2) s += qp[d2] * kp[d2];
  s = s * 0.125f + ((tok[b * SEQL + tid] != 0) ? 0.f : NEGF);
  ss[tid] = s; red[tid] = s;
  __syncthreads();
  for (int o = 128; o > 0; o >>= 1) {
    if (tid < o) red[tid] = fmaxf(red[tid], red[tid + o]);
    __syncthreads();
  }
  float mx = red[0];
  __syncthreads();
  float e = expf(ss[tid] - mx);
  ss[tid] = e; red[tid] = e;
  __syncthreads();
  for (int o = 128; o > 0; o >>= 1) {
    if (tid < o) red[tid] += red[tid + o];
    __syncthreads();
  }
  float sum = red[0];
  __syncthreads();
  ss[tid] = ss[tid] / sum;
  __syncthreads();
  int d2 = tid & 63, chunk = tid >> 6;
  float p = 0.f;
  for (int t = chunk * 64; t < chunk * 64 + 64; ++t)
    p += ss[t] * v[((size_t)(b * SEQL + t)) * DMODEL + h * 64 + d2];
  red[tid] = p;
  __syncthreads();
  if (tid < 64)
    ao[((size_t)(b * SEQL + qr)) * DMODEL + h * 64 + tid] =
        red[tid] + red[tid + 64] + red[tid + 128] + red[tid + 192];
}

// x = LayerNorm(x + r) * g + b   (one block of 128 per row)
__global__ void mek_add_ln(float* __restrict__ x, const float* __restrict__ r,
                           const float* __restrict__ g, const float* __restrict__ bb) {
  int row = blockIdx.x, tid = threadIdx.x;  // 128
  __shared__ float red[DMODEL];
  size_t idx = (size_t)row * DMODEL + tid;
  float val = x[idx] + r[idx];
  red[tid] = val;
  __syncthreads();
  for (int o = 64; o > 0; o >>= 1) {
    if (tid < o) red[tid] += red[tid + o];
    __syncthreads();
  }
  float mean = red[0] * (1.f / DMODEL);
  __syncthreads();
  float dv = val - mean;
  red[tid] = dv * dv;
  __syncthreads();
  for (int o = 64; o > 0; o >>= 1) {
    if (tid < o) red[tid] += red[tid + o];
    __syncthreads();
  }
  float var = red[0] * (1.f / DMODEL);
  x[idx] = dv * rsqrtf(var + 1e-5f) * g[tid] + bb[tid];
}

// ---------------------------------------------------------------- attention pooling
__global__ void mek_ent_pool(const float* __restrict__ kg, const int* __restrict__ ctx,
                             const float* __restrict__ Wa, const float* __restrict__ ba,
                             const float* __restrict__ vv, float* __restrict__ rep) {
  int b = blockIdx.x, tid = threadIdx.x;  // 256
  __shared__ float h[NCTX * DMODEL];
  __shared__ float red[256];
  __shared__ float sj[NCTX];
  for (int idx = tid; idx < NCTX * DMODEL; idx += 256) {
    int j = idx >> 7, d = idx & 127;
    int ent = ctx[b * NCTX + j];
    h[idx] = kg[(size_t)ent * DMODEL + d];
  }
  __syncthreads();
  int j = tid >> 3;
  int d0 = (tid & 7) * 16;
  float partial = 0.f;
  for (int d = d0; d < d0 + 16; ++d) {
    float acc = ba[d];
    for (int kk = 0; kk < DMODEL; ++kk) acc += h[j * DMODEL + kk] * Wa[kk * DMODEL + d];
    partial += vv[d] * tanhf(acc);
  }
  red[tid] = partial;
  __syncthreads();
  for (int o = 4; o > 0; o >>= 1) {
    if ((tid & 7) < o) red[tid] += red[tid + o];
    __syncthreads();
  }
  if ((tid & 7) == 0)
    sj[j] = (ctx[b * NCTX + j] != 0) ? red[tid] : NEGF;
  __syncthreads();
  if (tid == 0) {
    float mx = -3.4e38f;
    for (int jj = 0; jj < NCTX; ++jj) mx = fmaxf(mx, sj[jj]);
    float s = 0.f;
    for (int jj = 0; jj < NCTX; ++jj) { sj[jj] = expf(sj[jj] - mx); s += sj[jj]; }
    float inv = 1.f / s;
    for (int jj = 0; jj < NCTX; ++jj) sj[jj] *= inv;
  }
  __syncthreads();
  int d = tid & 127, half = tid >> 7;
  float p = 0.f;
  for (int jj = half * 16; jj < half * 16 + 16; ++jj) p += sj[jj] * h[jj * DMODEL + d];
  red[tid] = p;
  __syncthreads();
  if (tid < 128) rep[b * DMODEL + tid] = red[tid] + red[tid + 128];
}

__global__ void mek_tok_pool(const float* __restrict__ x, const int* __restrict__ tok,
                             const float* __restrict__ Wa, const float* __restrict__ ba,
                             const float* __restrict__ vv, float* __restrict__ rep) {
  int b = blockIdx.x, tid = threadIdx.x;  // 256, thread t owns row l=t
  __shared__ float ss[SEQL];
  __shared__ float red[SEQL];
  const float* xr = x + ((size_t)b * SEQL + tid) * DMODEL;
  float sacc = 0.f;
  for (int d = 0; d < DMODEL; ++d) {
    float acc = ba[d];
    for (int kk = 0; kk < DMODEL; ++kk) acc += xr[kk] * Wa[kk * DMODEL + d];
    sacc += vv[d] * tanhf(acc);
  }
  ss[tid] = (tok[b * SEQL + tid] != 0) ? sacc : NEGF;
  red[tid] = ss[tid];
  __syncthreads();
  for (int o = 128; o > 0; o >>= 1) {
    if (tid < o) red[tid] = fmaxf(red[tid], red[tid + o]);
    __syncthreads();
  }
  float mx = red[0];
  __syncthreads();
  float e = expf(ss[tid] - mx);
  ss[tid] = e; red[tid] = e;
  __syncthreads();
  for (int o = 128; o > 0; o >>= 1) {
    if (tid < o) red[tid] += red[tid + o];
    __syncthreads();
  }
  float sum = red[0];
  __syncthreads();
  ss[tid] = ss[tid] / sum;
  __syncthreads();
  int d = tid & 127, half = tid >> 7;
  float p = 0.f;
  for (int l = half * 128; l < half * 128 + 128; ++l)
    p += ss[l] * x[((size_t)b * SEQL + l) * DMODEL + d];
  red[tid] = p;
  __syncthreads();
  if (tid < 128) rep[b * DMODEL + tid] = red[tid] + red[tid + 128];
}

__global__ void mek_gate(const float* __restrict__ tr, const float* __restrict__ er,
                         const float* __restrict__ Wg, const float* __restrict__ bg,
                         float* __restrict__ user) {
  int b = blockIdx.x, d = threadIdx.x;  // 128
  float acc = bg[d];
  for (int kk = 0; kk < DMODEL; ++kk) acc += tr[b * DMODEL + kk] * Wg[kk * DMODEL + d];
  for (int kk = 0; kk < DMODEL; ++kk) acc += er[b * DMODEL + kk] * Wg[(DMODEL + kk) * DMODEL + d];
  float g = 1.f / (1.f + expf(-acc));
  user[b * DMODEL + d] = g * tr[b * DMODEL + d] + (1.f - g) * er[b * DMODEL + d];
}

// ---------------------------------------------------------------- out = user @ kg^T
// B fragment = kg rows (K contiguous) -> aligned pair loads.  30000 = 1875*16.
__global__ void mek_out_wmma(const float* __restrict__ user, const float* __restrict__ kg,
                             float* __restrict__ out) {
  int wave = threadIdx.x >> 5, lane = threadIdx.x & 31;
  int tile = blockIdx.x * 8 + wave;
  if (tile >= 4 * 1875) return;
  int mt = tile / 1875, nt = tile % 1875;
  int m = lane & 15, kh = lane >> 4;
  v8f c = {};
  const float* ap = user + (size_t)(mt * 16 + m) * DMODEL + 2 * kh;
  const float* bp = kg + (size_t)(nt * 16 + m) * DMODEL + 2 * kh;
  for (int kk = 0; kk < DMODEL; kk += 4) {
    v2f a, bf;
    a[0] = ap[kk];  a[1] = ap[kk + 1];
    bf[0] = bp[kk]; bf[1] = bp[kk + 1];
    c = __builtin_amdgcn_wmma_f32_16x16x4_f32(false, a, false, bf, (short)0, c, false, false);
  }
  for (int r2 = 0; r2 < 8; ++r2) {
    int row = mt * 16 + r2 + 8 * kh;
    out[(size_t)row * N_ENTITY + nt * 16 + m] = c[r2];
  }
}

// ---------------------------------------------------------------- host
extern "C" void kernel_launch(void* const* d_in, const int* in_sizes, int n_in,
                              void* d_out, int out_size, void* d_ws, size_t ws_size,
                              hipStream_t stream) {
  const int*   edge_idx  = (const int*)d_in[0];
  const int*   edge_type = (const int*)d_in[1];
  const int*   ctx_ent   = (const int*)d_in[2];
  const int*   ctx_tok   = (const int*)d_in[3];
  const float* basis     = (const float*)d_in[4];
  const float* comp      = (const float*)d_in[5];
  const float* root      = (const float*)d_in[6];
  const float* rgcn_bias = (const float*)d_in[7];
  const float* tok_emb   = (const float*)d_in[8];
  const float* pos_emb   = (const float*)d_in[9];
  const float* Wqkv      = (const float*)d_in[10];
  const float* Wo        = (const float*)d_in[11];
  const float* ln1_g     = (const float*)d_in[12];
  const float* ln1_b     = (const float*)d_in[13];
  const float* ln2_g     = (const float*)d_in[14];
  const float* ln2_b     = (const float*)d_in[15];
  const float* W1        = (const float*)d_in[16];
  const float* b1        = (const float*)d_in[17];
  const float* W2        = (const float*)d_in[18];
  const float* b2        = (const float*)d_in[19];
  const float* ent_Wa    = (const float*)d_in[20];
  const float* ent_ba    = (const float*)d_in[21];
  const float* ent_v     = (const float*)d_in[22];
  const float* tok_Wa    = (const float*)d_in[23];
  const float* tok_ba    = (const float*)d_in[24];
  const float* tok_v     = (const float*)d_in[25];
  const float* Wg        = (const float*)d_in[26];
  const float* bg        = (const float*)d_in[27];
  float* out = (float*)d_out;

  // workspace layout (floats)
  float* ws      = (float*)d_ws;
  float* cnt     = ws;                               // 540000
  float* kg      = cnt + (long)N_ENTITY * N_REL;     // 3840000
  float* x       = kg + (long)N_ENTITY * DMODEL;     // 2097152
  long   XSZ     = (long)BATCH * SEQL * DMODEL;
  float* q       = x + XSZ;
  float* kbuf    = q + XSZ;
  float* v       = kbuf + XSZ;
  float* ao      = v + XSZ;
  float* t1      = ao + XSZ;
  float* f       = t1 + XSZ;                         // 8388608
  float* ent_rep = f + (long)BATCH * SEQL * FFND;
  float* tok_rep = ent_rep + BATCH * DMODEL;
  float* user    = tok_rep + BATCH * DMODEL;

  // ---- RGCN ----
  long zn = (long)N_ENTITY * N_REL + (long)N_ENTITY * DMODEL;  // cnt + kg contiguous
  mek_zero<<<4096, 256, 0, stream>>>(cnt, zn);
  mek_edge_cnt<<<(NEDGE + 255) / 256, 256, 0, stream>>>(edge_idx, edge_type, cnt);
  mek_edge_msg<<<(NEDGE + 7) / 8, 256, 0, stream>>>(edge_idx, edge_type, basis, comp, cnt, kg);
  mek_kg_fin<<<(int)(((long)N_ENTITY * DMODEL + 255) / 256), 256, 0, stream>>>(kg, root, rgcn_bias);

  // ---- entity pooling ----
  mek_ent_pool<<<BATCH, 256, 0, stream>>>(kg, ctx_ent, ent_Wa, ent_ba, ent_v, ent_rep);

  // ---- token encoder ----
  mek_embed<<<(int)((XSZ + 255) / 256), 256, 0, stream>>>(ctx_tok, tok_emb, pos_emb, x);
  int M = BATCH * SEQL;                              // 16384
  int g128 = ((M / 16) * (DMODEL / 16) + 7) / 8;     // 1024 blocks
  int g512 = ((M / 16) * (FFND / 16) + 7) / 8;       // 4096 blocks
  for (int i = 0; i < 2; ++i) {
    const float* Wl = Wqkv + (size_t)i * 3 * DMODEL * DMODEL;
    mek_gemm<<<g128, 256, 0, stream>>>(x, Wl + 0 * DMODEL * DMODEL, q, nullptr, nullptr,
                                       M, DMODEL, DMODEL, DMODEL, DMODEL, DMODEL, 0);
    mek_gemm<<<g128, 256, 0, stream>>>(x, Wl + 1 * DMODEL * DMODEL, kbuf, nullptr, nullptr,
                                       M, DMODEL, DMODEL, DMODEL, DMODEL, DMODEL, 0);
    mek_gemm<<<g128, 256, 0, stream>>>(x, Wl + 2 * DMODEL * DMODEL, v, nullptr, nullptr,
                                       M, DMODEL, DMODEL, DMODEL, DMODEL, DMODEL, 0);
    mek_attn<<<BATCH * 2 * SEQL, 256, 0, stream>>>(q, kbuf, v, ctx_tok, ao);
    mek_gemm<<<g128, 256, 0, stream>>>(ao, Wo + (size_t)i * DMODEL * DMODEL, t1, nullptr, nullptr,
                                       M, DMODEL, DMODEL, DMODEL, DMODEL, DMODEL, 0);
    mek_add_ln<<<M, DMODEL, 0, stream>>>(x, t1, ln1_g + i * DMODEL, ln1_b + i * DMODEL);
    mek_gemm<<<g512, 256, 0, stream>>>(x, W1 + (size_t)i * DMODEL * FFND, f,
                                       b1 + i * FFND, nullptr,
                                       M, FFND, DMODEL, DMODEL, FFND, FFND, 1);
    mek_gemm<<<g128, 256, 0, stream>>>(f, W2 + (size_t)i * FFND * DMODEL, t1,
                                       b2 + i * DMODEL, nullptr,
                                       M, DMODEL, FFND, FFND, DMODEL, DMODEL, 0);
    mek_add_ln<<<M, DMODEL, 0, stream>>>(x, t1, ln2_g + i * DMODEL, ln2_b + i * DMODEL);
  }

  // ---- token pooling, gate, final score GEMM ----
  mek_tok_pool<<<BATCH, 256, 0, stream>>>(x, ctx_tok, tok_Wa, tok_ba, tok_v, tok_rep);
  mek_gate<<<BATCH, DMODEL, 0, stream>>>(tok_rep, ent_rep, Wg, bg, user);
  mek_out_wmma<<<(4 * 1875 + 7) / 8, 256, 0, stream>>>(user, kg, out);
}